// TemporalBatchTopKSAE_35931696399070
// MI455X (gfx1250) — compile-verified
//
#include <hip/hip_runtime.h>

typedef __attribute__((ext_vector_type(2))) float v2f;
typedef __attribute__((ext_vector_type(8))) float v8f;

#define B_SZ   4096
#define D_IN   2048
#define D_SAE  16384
#define K_TOP  64
#define K_TOTAL (B_SZ * K_TOP)        // 262144

// ---------------- Encoder GEMM: pre = relu(x @ W_enc + beff) ----------------
// beff[n] = b_enc[n] - sum_k b_dec[k] * W_enc[k][n]   (precomputed)
#define MT 128
#define NT 128
#define KT 32
#define AS_STRIDE 36                  // 32+4 pad floats; 36*4=144 B rows, 16B aligned
#define BS_STRIDE 132                 // 128+4 pad floats; 132*4=528 B rows, 16B aligned
#define AS_FLOATS (MT * AS_STRIDE)    // 4608
#define BS_FLOATS (KT * BS_STRIDE)    // 4224
#define BUF_FLOATS (AS_FLOATS + BS_FLOATS)

#if defined(__has_builtin) && __has_builtin(__builtin_amdgcn_s_wait_asynccnt)
#define WAIT_ASYNC(n) __builtin_amdgcn_s_wait_asynccnt(n)
#else
#define WAIT_ASYNC(n) asm volatile("s_wait_asynccnt %0" :: "i"(n) : "memory")
#endif

__device__ __forceinline__ void async_cp16(float* lds_dst, const float* gsrc) {
    // CDNA5 async DMA: global -> LDS, 16B, no VGPR data path (ASYNCcnt tracked)
    asm volatile("global_load_async_to_lds_b128 %0, %1, off"
                 :: "v"((unsigned)(uintptr_t)lds_dst),
                    "v"((unsigned long long)(uintptr_t)gsrc)
                 : "memory");
}

__device__ __forceinline__ void stage_tile(const float* __restrict__ x,
                                           const float* __restrict__ W_enc,
                                           float* As, float* Bs,
                                           int row0, int n0, int kk, int tid)
{
    // x tile: 128x32 fp32 = 1024 x b128 -> 4 async ops/lane
    for (int i = 0; i < 4; ++i) {
        int c  = tid + i * 256;
        int m  = c >> 3;
        int k4 = c & 7;
        async_cp16(As + m * AS_STRIDE + k4 * 4,
                   x + (size_t)(row0 + m) * D_IN + kk + k4 * 4);
    }
    // W tile: 32x128 fp32 = 1024 x b128 -> 4 async ops/lane
    for (int i = 0; i < 4; ++i) {
        int c  = tid + i * 256;
        int k  = c >> 5;
        int n4 = c & 31;
        async_cp16(Bs + k * BS_STRIDE + n4 * 4,
                   W_enc + (size_t)(kk + k) * D_SAE + n0 + n4 * 4);
    }
}

__global__ __launch_bounds__(256) void sae_enc_gemm(
    const float* __restrict__ x, const float* __restrict__ W_enc,
    const float* __restrict__ beff, float* __restrict__ zbuf)
{
    __shared__ float lds[2][BUF_FLOATS];   // double-buffered {As | Bs}

    const int tid  = threadIdx.x;
    const int lane = tid & 31;
    const int wave = tid >> 5;
    const int hi   = lane >> 4;            // half-wave: 0 or 1
    const int lml  = lane & 15;

    const int n0   = blockIdx.x * NT;
    const int row0 = blockIdx.y * MT;

    const int m_wave = (wave & 1) * 64;    // 2x4 wave grid: 64x32 per wave
    const int n_wave = (wave >> 1) * 32;

    v8f acc[4][2];
    for (int i = 0; i < 4; ++i)
        for (int j = 0; j < 2; ++j)
            for (int r = 0; r < 8; ++r) acc[i][j][r] = 0.0f;

    const int NK = D_IN / KT;              // 64 K-tiles

    stage_tile(x, W_enc, lds[0], lds[0] + AS_FLOATS, row0, n0, 0, tid);

    for (int it = 0; it < NK; ++it) {
        const int cur = it & 1;
        if (it + 1 < NK) {
            stage_tile(x, W_enc, lds[cur ^ 1], lds[cur ^ 1] + AS_FLOATS,
                       row0, n0, (it + 1) * KT, tid);
            WAIT_ASYNC(8);                 // in-order: previous stage (8 ops) landed
        } else {
            WAIT_ASYNC(0);
        }
        __syncthreads();

        const float* As = lds[cur];
        const float* Bs = lds[cur] + AS_FLOATS;

        for (int k0 = 0; k0 < KT; k0 += 4) {
            const int ka = k0 + hi * 2;    // lanes 0-15: K=k0,k0+1 ; lanes 16-31: K=k0+2,k0+3
            v2f a[4], b[2];
            for (int mf = 0; mf < 4; ++mf) {
                const float* p = As + (m_wave + mf * 16 + lml) * AS_STRIDE + ka;
                a[mf] = *(const v2f*)p;
            }
            for (int nf = 0; nf < 2; ++nf) {
                const float* p = Bs + ka * BS_STRIDE + n_wave + nf * 16 + lml;
                v2f t; t[0] = p[0]; t[1] = p[BS_STRIDE];
                b[nf] = t;
            }
            for (int mf = 0; mf < 4; ++mf)
                for (int nf = 0; nf < 2; ++nf)
                    acc[mf][nf] = __builtin_amdgcn_wmma_f32_16x16x4_f32(
                        false, a[mf], false, b[nf], (short)0, acc[mf][nf], false, false);
        }
        __syncthreads();                   // buffer reuse fence
    }

    // epilogue: relu(acc + beff) -> z buffer (pre-activations)
    for (int nf = 0; nf < 2; ++nf) {
        const int col = n0 + n_wave + nf * 16 + lml;
        const float be = beff[col];
        for (int mf = 0; mf < 4; ++mf) {
            for (int r = 0; r < 8; ++r) {
                const int row = row0 + m_wave + mf * 16 + r + hi * 8;
                float v = acc[mf][nf][r] + be;
                zbuf[(size_t)row * D_SAE + col] = v > 0.0f ? v : 0.0f;
            }
        }
    }
}

// ---------------- bias fold: beff = b_enc - b_dec @ W_enc -------------------
__global__ __launch_bounds__(256) void enc_bias_fold(
    const float* __restrict__ W_enc, const float* __restrict__ b_enc,
    const float* __restrict__ b_dec, float* __restrict__ beff)
{
    const int n = blockIdx.x * 256 + threadIdx.x;
    float s = 0.0f;
    for (int k = 0; k < D_IN; ++k)
        s += b_dec[k] * W_enc[(size_t)k * D_SAE + n];
    beff[n] = b_enc[n] - s;
}

// ---------------- Exact global top-k threshold via 32-bit radix select ------
// ws u32 layout: [0..8191]=hist, [8192]=k_rem, [8193]=prefix, [8194]=mask,
//                [8195]=threshold, [8448..8448+16383]=beff (float)
#define NBINS 8192
#define WS_BEFF_OFF 8448

__global__ void topk_init(unsigned* ws) {
    int t = blockIdx.x * blockDim.x + threadIdx.x;
    if (t < NBINS) ws[t] = 0;
    if (t == 0) { ws[NBINS] = K_TOTAL; ws[NBINS+1] = 0; ws[NBINS+2] = 0; ws[NBINS+3] = 0; }
}

__global__ __launch_bounds__(256) void topk_hist(
    const float* __restrict__ zbuf, unsigned* __restrict__ ws, int shift, int nbins)
{
    __shared__ unsigned h[NBINS];
    for (int i = threadIdx.x; i < nbins; i += 256) h[i] = 0;
    __syncthreads();
    const unsigned prefix  = ws[NBINS + 1];
    const unsigned pmask   = ws[NBINS + 2];
    const unsigned binmask = (unsigned)(nbins - 1);
    const size_t n4 = (size_t)B_SZ * D_SAE / 4;
    const size_t stride = (size_t)gridDim.x * blockDim.x;
    for (size_t i = (size_t)blockIdx.x * blockDim.x + threadIdx.x; i < n4; i += stride) {
        float4 v = ((const float4*)zbuf)[i];
        unsigned b;
        b = __float_as_uint(v.x); if ((b & pmask) == prefix) atomicAdd(&h[(b >> shift) & binmask], 1u);
        b = __float_as_uint(v.y); if ((b & pmask) == prefix) atomicAdd(&h[(b >> shift) & binmask], 1u);
        b = __float_as_uint(v.z); if ((b & pmask) == prefix) atomicAdd(&h[(b >> shift) & binmask], 1u);
        b = __float_as_uint(v.w); if ((b & pmask) == prefix) atomicAdd(&h[(b >> shift) & binmask], 1u);
    }
    __syncthreads();
    for (int i = threadIdx.x; i < nbins; i += 256) {
        unsigned c = h[i];
        if (c) atomicAdd(&ws[i], c);
    }
}

__global__ void topk_select(unsigned* ws, int shift, int nbins) {
    if (threadIdx.x == 0) {
        unsigned k_rem = ws[NBINS];
        unsigned cum = 0;
        int chosen = 0;
        for (int b = nbins - 1; b >= 0; --b) {
            unsigned c = ws[b];
            if (cum + c >= k_rem) { chosen = b; k_rem = k_rem - cum; break; }
            cum += c;
        }
        unsigned newpref = ws[NBINS + 1] | ((unsigned)chosen << shift);
        ws[NBINS]     = k_rem;
        ws[NBINS + 1] = newpref;
        ws[NBINS + 2] = ws[NBINS + 2] | (((unsigned)(nbins - 1)) << shift);
        ws[NBINS + 3] = newpref;   // valid threshold after the final level
    }
    __syncthreads();
    for (int i = threadIdx.x; i < NBINS; i += blockDim.x) ws[i] = 0;  // reset for next level
}

// ---------------- Fused mask + sparse decoder: x_hat = z @ W_dec + b_dec ----
#define DCHUNK 2048
#define SEG (DCHUNK / 256)   // 8 contiguous features per thread -> deterministic order

__global__ __launch_bounds__(256) void sae_decode(
    float* __restrict__ zbuf, const float* __restrict__ W_dec,
    const float* __restrict__ b_dec, float* __restrict__ xhat,
    const unsigned* __restrict__ ws)
{
    __shared__ unsigned s_idx[DCHUNK];
    __shared__ float    s_val[DCHUNK];
    __shared__ unsigned s_cnt[257];

    const int tid = threadIdx.x;
    const int row = blockIdx.x;
    const float thr = __uint_as_float(ws[NBINS + 3]);
    float* zrow = zbuf + (size_t)row * D_SAE;

    float acc[8];
    for (int j = 0; j < 8; ++j) acc[j] = 0.0f;

    for (int c0 = 0; c0 < D_SAE; c0 += DCHUNK) {
        float vloc[SEG];
        unsigned cnt = 0;
        const int fbase = c0 + tid * SEG;
        for (int j = 0; j < SEG; ++j) {
            float v = zrow[fbase + j];
            bool keep = (v >= thr) && (v > 0.0f);
            float m = keep ? v : 0.0f;
            zrow[fbase + j] = m;        // write masked z in place (fully rewritten every call)
            vloc[j] = m;
            cnt += keep ? 1u : 0u;
        }
        s_cnt[tid] = cnt;
        __syncthreads();
        if (tid == 0) {
            unsigned run = 0;
            for (int i = 0; i < 256; ++i) { unsigned c = s_cnt[i]; s_cnt[i] = run; run += c; }
            s_cnt[256] = run;
        }
        __syncthreads();
        unsigned off = s_cnt[tid];
        for (int j = 0; j < SEG; ++j) {
            if (vloc[j] > 0.0f) { s_idx[off] = (unsigned)(fbase + j); s_val[off] = vloc[j]; ++off; }
        }
        __syncthreads();
        const unsigned n = s_cnt[256];
        for (unsigned e = 0; e < n; ++e) {           // index-ordered -> deterministic fp sum
            const float val = s_val[e];
            const float4* wr = (const float4*)(W_dec + (size_t)s_idx[e] * D_IN) + tid * 2;
            float4 w0 = wr[0], w1 = wr[1];
            acc[0] += val * w0.x; acc[1] += val * w0.y; acc[2] += val * w0.z; acc[3] += val * w0.w;
            acc[4] += val * w1.x; acc[5] += val * w1.y; acc[6] += val * w1.z; acc[7] += val * w1.w;
        }
        __syncthreads();
    }

    float* xr = xhat + (size_t)row * D_IN + tid * 8;
    const float* bd = b_dec + tid * 8;
    float4 o0 = make_float4(acc[0] + bd[0], acc[1] + bd[1], acc[2] + bd[2], acc[3] + bd[3]);
    float4 o1 = make_float4(acc[4] + bd[4], acc[5] + bd[5], acc[6] + bd[6], acc[7] + bd[7]);
    *(float4*)xr = o0;
    *((float4*)xr + 1) = o1;
}

// ---------------------------------------------------------------------------
extern "C" void kernel_launch(void* const* d_in, const int* in_sizes, int n_in,
                              void* d_out, int out_size, void* d_ws, size_t ws_size,
                              hipStream_t stream) {
    const float* x     = (const float*)d_in[0];
    const float* W_enc = (const float*)d_in[1];
    const float* b_enc = (const float*)d_in[2];
    const float* W_dec = (const float*)d_in[3];
    const float* b_dec = (const float*)d_in[4];

    float* xhat = (float*)d_out;                         // (B, D_IN)
    float* zbuf = xhat + (size_t)B_SZ * D_IN;            // (B, D_SAE), second output
    unsigned* ws = (unsigned*)d_ws;
    float* beff = (float*)(ws + WS_BEFF_OFF);            // (D_SAE,)

    enc_bias_fold<<<dim3(D_SAE / 256), dim3(256), 0, stream>>>(W_enc, b_enc, b_dec, beff);

    sae_enc_gemm<<<dim3(D_SAE / NT, B_SZ / MT), dim3(256), 0, stream>>>(
        x, W_enc, beff, zbuf);

    topk_init<<<dim3(32), dim3(256), 0, stream>>>(ws);
    // radix select on positive-float bit patterns: 13 + 13 + 6 bits
    topk_hist<<<dim3(2048), dim3(256), 0, stream>>>(zbuf, ws, 19, 8192);
    topk_select<<<dim3(1), dim3(256), 0, stream>>>(ws, 19, 8192);
    topk_hist<<<dim3(2048), dim3(256), 0, stream>>>(zbuf, ws, 6, 8192);
    topk_select<<<dim3(1), dim3(256), 0, stream>>>(ws, 6, 8192);
    topk_hist<<<dim3(2048), dim3(256), 0, stream>>>(zbuf, ws, 0, 64);
    topk_select<<<dim3(1), dim3(256), 0, stream>>>(ws, 0, 64);

    sae_decode<<<dim3(B_SZ), dim3(256), 0, stream>>>(zbuf, W_dec, b_dec, xhat, ws);
}